// RTFMVDR_19026705121951
// MI455X (gfx1250) — compile-verified
//
#include <hip/hip_runtime.h>
#include <hip/hip_bf16.h>

// Problem constants from the reference
#define BB 8
#define CC 8
#define FF 513
#define TT 1500
#define NTILE ((TT + 15) / 16)   // 94 time tiles of 16

typedef float v2f __attribute__((ext_vector_type(2)));
typedef float v8f __attribute__((ext_vector_type(8)));

// ---------------------------------------------------------------------------
// Kernel 1: per-(b,f) MVDR weight solve.
// One thread per (b,f): 8x8 complex Gaussian elimination (no pivoting: psd is
// Hermitian PSD + Tikhonov loading => positive definite), fully unrolled so
// the matrix lives in VGPRs. Outputs w[b,f,c] (complex interleaved) to d_ws.
// ---------------------------------------------------------------------------
__global__ void mvdr_solve_kernel(const float* __restrict__ psd,   // (B,F,C,C) cplx
                                  const float* __restrict__ rtf,   // (B,F,C) cplx
                                  const int*   __restrict__ refch,
                                  float*       __restrict__ w_out, // (B*F,8) cplx
                                  int BF) {
    int tid = blockIdx.x * blockDim.x + threadIdx.x;
    if (tid >= BF) return;

    const float* A = psd + (size_t)tid * (CC * CC * 2);
    const float* R = rtf + (size_t)tid * (CC * 2);

    float ar[8][8], ai[8][8], xr[8], xi[8], rr[8], ri[8];
    float tr = 0.f;
#pragma unroll
    for (int i = 0; i < 8; ++i) {
#pragma unroll
        for (int j = 0; j < 8; ++j) {
            ar[i][j] = A[(i * 8 + j) * 2 + 0];
            ai[i][j] = A[(i * 8 + j) * 2 + 1];
        }
        tr += ar[i][i];
        rr[i] = R[2 * i];
        ri[i] = R[2 * i + 1];
        xr[i] = rr[i];
        xi[i] = ri[i];
    }
    // Tikhonov loading: eps = trace.real * 1e-7 + 1e-8 added to the diagonal
    float epsv = tr * 1e-7f + 1e-8f;
#pragma unroll
    for (int i = 0; i < 8; ++i) ar[i][i] += epsv;

    // Forward elimination (row k scaled so diag becomes 1)
#pragma unroll
    for (int k = 0; k < 8; ++k) {
        float dr = ar[k][k], di = ai[k][k];
        float s  = 1.f / (dr * dr + di * di);
        float pr = dr * s, pi = -di * s;            // 1 / a[k][k]
#pragma unroll
        for (int j = k + 1; j < 8; ++j) {
            float tr_ = ar[k][j] * pr - ai[k][j] * pi;
            float ti_ = ar[k][j] * pi + ai[k][j] * pr;
            ar[k][j] = tr_; ai[k][j] = ti_;
        }
        {
            float tr_ = xr[k] * pr - xi[k] * pi;
            float ti_ = xr[k] * pi + xi[k] * pr;
            xr[k] = tr_; xi[k] = ti_;
        }
#pragma unroll
        for (int i = k + 1; i < 8; ++i) {
            float mr = ar[i][k], mi = ai[i][k];
#pragma unroll
            for (int j = k + 1; j < 8; ++j) {
                ar[i][j] -= mr * ar[k][j] - mi * ai[k][j];
                ai[i][j] -= mr * ai[k][j] + mi * ar[k][j];
            }
            xr[i] -= mr * xr[k] - mi * xi[k];
            xi[i] -= mr * xi[k] + mi * xr[k];
        }
    }
    // Back substitution (unit diagonal)
#pragma unroll
    for (int k = 7; k >= 0; --k) {
#pragma unroll
        for (int j = k + 1; j < 8; ++j) {
            xr[k] -= ar[k][j] * xr[j] - ai[k][j] * xi[j];
            xi[k] -= ar[k][j] * xi[j] + ai[k][j] * xr[j];
        }
    }

    // denominator = Re(rtf^H x) + 1e-8
    float den = 1e-8f;
#pragma unroll
    for (int i = 0; i < 8; ++i) den += rr[i] * xr[i] + ri[i] * xi[i];
    float inv = 1.f / den;

    // reference-channel rescale (select without dynamic register indexing)
    int rc = refch[0] & 7;
    float cr = 0.f, ci = 0.f;
#pragma unroll
    for (int i = 0; i < 8; ++i)
        if (i == rc) { cr = rr[i]; ci = ri[i]; }

    float* W = w_out + (size_t)tid * 16;
#pragma unroll
    for (int i = 0; i < 8; ++i) {
        W[2 * i + 0] = (xr[i] * cr - xi[i] * ci) * inv;
        W[2 * i + 1] = (xr[i] * ci + xi[i] * cr) * inv;
    }
}

// ---------------------------------------------------------------------------
// Kernel 2: beamforming apply via V_WMMA_F32_16X16X4_F32 (fp32 precision).
//   out[t] = sum_c conj(w[c]) * y[c,t]
// Real GEMM with K=16, re/im interleaved along K so each loaded {re,im} b64
// pair IS one even-aligned A fragment (no repacking movs):
//   A[t, 2c+0] = yr[c,t],  A[t, 2c+1] = yi[c,t]
//   B[2c+0, 0] =  wr[c],   B[2c+1, 0] = wi[c]    (out_re = wr*yr + wi*yi)
//   B[2c+0, 1] = -wi[c],   B[2c+1, 1] = wr[c]    (out_im = wr*yi - wi*yr)
// One block (8 waves) per (b,f); waves stripe 16-time tiles, chaining four
// K=4 WMMAs per tile. Non-temporal loads/stores: 394 MB streamed exactly once
// (no reuse; don't thrash the 192 MB L2). Loads use clamped addresses so EXEC
// stays all-ones around the WMMAs; the tail check is wave-uniform and only
// the last tile takes the predicated store path.
// ---------------------------------------------------------------------------
__global__ __launch_bounds__(256)
void mvdr_apply_kernel(const float* __restrict__ spec,  // (B,C,F,T) cplx
                       const float* __restrict__ w_ws,  // (B*F,8) cplx
                       float*       __restrict__ out) { // (B,F,T) cplx
    const int bf   = blockIdx.x;
    const int b    = bf / FF;
    const int f    = bf - b * FF;
    const int lane = threadIdx.x & 31;
    const int wave = threadIdx.x >> 5;

    const int lane15 = lane & 15;
    const int khalf  = (lane >> 4) << 1;   // +0 (lanes 0-15) or +2 (lanes 16-31)

    // ---- Build the 4 B fragments (constant over the whole time loop) ----
    const float* wv = w_ws + (size_t)bf * 16;
    v2f bfrag[4];
#pragma unroll
    for (int j = 0; j < 4; ++j) {
#pragma unroll
        for (int vg = 0; vg < 2; ++vg) {
            int k    = 4 * j + khalf + vg; // global K index 0..15
            int c    = k >> 1;             // channel
            int part = k & 1;              // 0 = yr term, 1 = yi term
            float wr = wv[2 * c + 0];
            float wi = wv[2 * c + 1];
            float val = 0.f;
            if (lane15 == 0)      val = part ? wi : wr;    // N=0 -> out_re
            else if (lane15 == 1) val = part ? wr : -wi;   // N=1 -> out_im
            bfrag[j][vg] = val;
        }
    }

    // ---- Per-lane channel base offsets (complex element base) ----
    // WMMA j's A fragment = {yr,yi} of channel c = 2j + (laneHi ? 1 : 0).
    long long cbase[4];
#pragma unroll
    for (int j = 0; j < 4; ++j) {
        int c = 2 * j + (khalf >> 1);      // channel 0..7
        cbase[j] = 2LL * ((((long long)b * CC + c) * FF + f) * TT);
    }

    const long long obase = 2LL * (((long long)b * FF + f) * TT);
    const int mb = (lane >> 4) * 8;        // output row half for C/D layout

    // ---- Time-tile loop: waves stripe tiles of 16 samples ----
    for (int tile = wave; tile < NTILE; tile += 8) {
        const int t0 = tile * 16;
        int tc = t0 + lane15;              // A row (time); clamp so loads stay
        if (tc > TT - 1) tc = TT - 1;      // in-bounds with EXEC all-ones

        v2f a[4];
#pragma unroll
        for (int j = 0; j < 4; ++j)        // one NT b64 = one A fragment
            a[j] = __builtin_nontemporal_load(
                (const v2f*)(spec + cbase[j] + 2LL * tc));

        v8f acc = {0.f, 0.f, 0.f, 0.f, 0.f, 0.f, 0.f, 0.f};
        acc = __builtin_amdgcn_wmma_f32_16x16x4_f32(false, a[0], false, bfrag[0],
                                                    (short)0, acc, false, false);
        acc = __builtin_amdgcn_wmma_f32_16x16x4_f32(false, a[1], false, bfrag[1],
                                                    (short)0, acc, false, false);
        acc = __builtin_amdgcn_wmma_f32_16x16x4_f32(false, a[2], false, bfrag[2],
                                                    (short)0, acc, false, false);
        acc = __builtin_amdgcn_wmma_f32_16x16x4_f32(false, a[3], false, bfrag[3],
                                                    (short)0, acc, false, false);

        // Columns 0/1 of D hold re/im; lanes {0,1,16,17} write the results.
        // Full-tile check is wave-uniform: only the last tile is predicated.
        if (lane15 < 2) {
            const long long p = obase + 2LL * (t0 + mb) + lane15;
            if (t0 + 16 <= TT) {
#pragma unroll
                for (int v = 0; v < 8; ++v)
                    __builtin_nontemporal_store(acc[v], out + p + 2LL * v);
            } else {
#pragma unroll
                for (int v = 0; v < 8; ++v)
                    if (t0 + mb + v < TT)
                        __builtin_nontemporal_store(acc[v], out + p + 2LL * v);
            }
        }
    }
}

// ---------------------------------------------------------------------------
extern "C" void kernel_launch(void* const* d_in, const int* in_sizes, int n_in,
                              void* d_out, int out_size, void* d_ws, size_t ws_size,
                              hipStream_t stream) {
    const float* spec = (const float*)d_in[0];   // specgram complex64
    const float* rtf  = (const float*)d_in[1];   // rtf complex64
    const float* psd  = (const float*)d_in[2];   // psd_n complex64
    const int*   refc = (const int*)d_in[3];     // reference_channel
    float* w_ws = (float*)d_ws;                  // 4104 * 16 floats = 262 KB
    float* outp = (float*)d_out;

    const int BF = BB * FF;                      // 4104

    mvdr_solve_kernel<<<(BF + 63) / 64, 64, 0, stream>>>(psd, rtf, refc, w_ws, BF);
    mvdr_apply_kernel<<<BF, 256, 0, stream>>>(spec, w_ws, outp);
}